// MyModel_83597243450144
// MI455X (gfx1250) — compile-verified
//
#include <hip/hip_runtime.h>

// LSTM autoencoder: B=2048, T=128, H1=128, H2=64.
// One workgroup (256 thr = 8 waves) owns a 16-row batch tile, keeps h/c in LDS,
// iterates the recurrence. Gate GEMMs use V_WMMA_F32_16X16X4_F32 (full f32).

#define BATCH 2048
#define TSTEPS 128
#define HH1 128
#define HH2 64

typedef float v2f __attribute__((ext_vector_type(2)));
typedef float v8f __attribute__((ext_vector_type(8)));

// One 16x16 output tile of  acc += A[16,K] @ B[K, ldB]  (columns noff..noff+15).
// A in LDS (row stride ldA), B in global (row stride ldB).
// l15 = lane&15 (row of A / col of B), kk = (lane>=16 ? 2 : 0).
__device__ __forceinline__ v8f mm16(v8f acc, const float* __restrict__ A, int ldA, int K,
                                    const float* __restrict__ B, int ldB, int noff,
                                    int l15, int kk) {
#pragma unroll 4
  for (int k0 = 0; k0 < K; k0 += 4) {
    int ka = k0 + kk;
    v2f a, b;
    a.x = A[l15 * ldA + ka];
    a.y = A[l15 * ldA + ka + 1];
    b.x = B[ka * ldB + noff + l15];
    b.y = B[(ka + 1) * ldB + noff + l15];
    acc = __builtin_amdgcn_wmma_f32_16x16x4_f32(false, a, false, b, (short)0, acc,
                                                false, false);
  }
  return acc;
}

// Elementwise LSTM cell update: z[16,4H] (LDS, stride LDZ_) -> h,c[16,H] (stride LDH_).
// Gate order i,f,g,o; g=relu, h = o * relu(c).
template <int H, int LDZ_, int LDH_>
__device__ __forceinline__ void lstm_update(const float* __restrict__ z,
                                            float* __restrict__ h,
                                            float* __restrict__ c, int tid) {
  for (int e = tid; e < 16 * H; e += 256) {
    int m = e / H;
    int q = e - m * H;
    const float* zr = z + m * LDZ_;
    float zi = zr[q];
    float zf = zr[H + q];
    float zg = zr[2 * H + q];
    float zo = zr[3 * H + q];
    float ig = 1.f / (1.f + __expf(-zi));
    float fg = 1.f / (1.f + __expf(-zf));
    float og = 1.f / (1.f + __expf(-zo));
    float g = zg > 0.f ? zg : 0.f;
    float cn = fg * c[m * LDH_ + q] + ig * g;
    c[m * LDH_ + q] = cn;
    h[m * LDH_ + q] = og * (cn > 0.f ? cn : 0.f);
  }
}

__global__ __launch_bounds__(256) void lstm_ae_kernel(
    const float* __restrict__ x,
    const float* __restrict__ W1, const float* __restrict__ U1, const float* __restrict__ b1,
    const float* __restrict__ W2, const float* __restrict__ U2, const float* __restrict__ b2,
    const float* __restrict__ W3, const float* __restrict__ U3, const float* __restrict__ b3,
    const float* __restrict__ W4, const float* __restrict__ U4, const float* __restrict__ b4,
    const float* __restrict__ Wd, const float* __restrict__ bd,
    float* __restrict__ out) {
  constexpr int LD1 = 132;   // 128 + 4 pad (stride%64 != 0 -> no LDS bank conflicts)
  constexpr int LD2 = 68;    // 64 + 4 pad
  constexpr int LDZ = 516;   // 512 + 4 pad (gate buffer)
  constexpr int LDZ3 = 260;  // 256 + 4 pad (precomputed layer-3 input term)
  constexpr int LDX = 129;

  __shared__ float h1[16][LD1], c1[16][LD1];
  __shared__ float h2[16][LD2], c2[16][LD2];
  __shared__ float h3[16][LD2], c3[16][LD2];
  __shared__ float h4[16][LD1], c4[16][LD1];
  __shared__ float zb[16][LDZ];
  __shared__ float zx3[16][LDZ3];
  __shared__ float xt[16][LDX];
  __shared__ float dred[16][8];

  const int tid = threadIdx.x;
  const int lane = tid & 31;
  const int wave = tid >> 5;
  const int l15 = lane & 15;
  const int kk = (lane >> 4) << 1;  // A/B k sub-offset: 0 or 2
  const int mb = (lane >> 4) << 3;  // C/D row base: 0 or 8
  const int b0 = blockIdx.x * 16;

  // Zero recurrent state, preload x tile [16][T] (coalesced).
  for (int i = tid; i < 16 * LD1; i += 256) {
    (&h1[0][0])[i] = 0.f; (&c1[0][0])[i] = 0.f;
    (&h4[0][0])[i] = 0.f; (&c4[0][0])[i] = 0.f;
  }
  for (int i = tid; i < 16 * LD2; i += 256) {
    (&h2[0][0])[i] = 0.f; (&c2[0][0])[i] = 0.f;
    (&h3[0][0])[i] = 0.f; (&c3[0][0])[i] = 0.f;
  }
  for (int i = tid; i < 16 * TSTEPS; i += 256) {
    int m = i >> 7, t = i & (TSTEPS - 1);
    xt[m][t] = x[(size_t)(b0 + m) * TSTEPS + t];
  }
  __syncthreads();

  // ---------------- Phase A: layers 1-2, fused per timestep ----------------
  for (int t = 0; t < TSTEPS; ++t) {
    // Layer 1: z = x_t*W1 + h1@U1 + b1   (4H1 = 512 cols -> 4 tiles/wave)
    for (int j = 0; j < 4; ++j) {
      int noff = (wave + 8 * j) * 16;
      float bv = b1[noff + l15];
      float wv = W1[noff + l15];  // W1 is [1, 512]
      v8f acc;
#pragma unroll
      for (int r = 0; r < 8; ++r) acc[r] = bv + xt[mb + r][t] * wv;
      acc = mm16(acc, &h1[0][0], LD1, HH1, U1, 4 * HH1, noff, l15, kk);
#pragma unroll
      for (int r = 0; r < 8; ++r) zb[mb + r][noff + l15] = acc[r];
    }
    __syncthreads();
    lstm_update<HH1, LDZ, LD1>(&zb[0][0], &h1[0][0], &c1[0][0], tid);
    __syncthreads();

    // Layer 2: z = h1@W2 + h2@U2 + b2   (4H2 = 256 cols -> 2 tiles/wave)
    for (int j = 0; j < 2; ++j) {
      int noff = (wave + 8 * j) * 16;
      float bv = b2[noff + l15];
      v8f acc;
#pragma unroll
      for (int r = 0; r < 8; ++r) acc[r] = bv;
      acc = mm16(acc, &h1[0][0], LD1, HH1, W2, 4 * HH2, noff, l15, kk);
      acc = mm16(acc, &h2[0][0], LD2, HH2, U2, 4 * HH2, noff, l15, kk);
#pragma unroll
      for (int r = 0; r < 8; ++r) zb[mb + r][noff + l15] = acc[r];
    }
    __syncthreads();
    lstm_update<HH2, LDZ, LD2>(&zb[0][0], &h2[0][0], &c2[0][0], tid);
    __syncthreads();
  }

  // RepeatVector: layer-3 input is h2_final for every t -> precompute once.
  for (int j = 0; j < 2; ++j) {
    int noff = (wave + 8 * j) * 16;
    float bv = b3[noff + l15];
    v8f acc;
#pragma unroll
    for (int r = 0; r < 8; ++r) acc[r] = bv;
    acc = mm16(acc, &h2[0][0], LD2, HH2, W3, 4 * HH2, noff, l15, kk);
#pragma unroll
    for (int r = 0; r < 8; ++r) zx3[mb + r][noff + l15] = acc[r];
  }
  __syncthreads();

  // ---------------- Phase B: layers 3-4 + dense, per timestep ----------------
  for (int t = 0; t < TSTEPS; ++t) {
    // Layer 3: z = zx3 + h3@U3
    for (int j = 0; j < 2; ++j) {
      int noff = (wave + 8 * j) * 16;
      v8f acc;
#pragma unroll
      for (int r = 0; r < 8; ++r) acc[r] = zx3[mb + r][noff + l15];
      acc = mm16(acc, &h3[0][0], LD2, HH2, U3, 4 * HH2, noff, l15, kk);
#pragma unroll
      for (int r = 0; r < 8; ++r) zb[mb + r][noff + l15] = acc[r];
    }
    __syncthreads();
    lstm_update<HH2, LDZ, LD2>(&zb[0][0], &h3[0][0], &c3[0][0], tid);
    __syncthreads();

    // Layer 4: z = h3@W4 + h4@U4 + b4   (4H1 = 512 cols -> 4 tiles/wave)
    for (int j = 0; j < 4; ++j) {
      int noff = (wave + 8 * j) * 16;
      float bv = b4[noff + l15];
      v8f acc;
#pragma unroll
      for (int r = 0; r < 8; ++r) acc[r] = bv;
      acc = mm16(acc, &h3[0][0], LD2, HH2, W4, 4 * HH1, noff, l15, kk);
      acc = mm16(acc, &h4[0][0], LD1, HH1, U4, 4 * HH1, noff, l15, kk);
#pragma unroll
      for (int r = 0; r < 8; ++r) zb[mb + r][noff + l15] = acc[r];
    }
    __syncthreads();
    lstm_update<HH1, LDZ, LD1>(&zb[0][0], &h4[0][0], &c4[0][0], tid);
    __syncthreads();

    // TimeDistributed Dense(1): out[b,t] = h4[m,:]@Wd + bd
    if (tid < 128) {
      int m = tid >> 3, seg = tid & 7;
      float s = 0.f;
#pragma unroll
      for (int k = 0; k < 16; ++k) s += h4[m][seg * 16 + k] * Wd[seg * 16 + k];
      dred[m][seg] = s;
    }
    __syncthreads();
    if (tid < 16) {
      float s = bd[0];
#pragma unroll
      for (int j = 0; j < 8; ++j) s += dred[tid][j];
      out[(size_t)(b0 + tid) * TSTEPS + t] = s;
    }
    __syncthreads();
  }
}

extern "C" void kernel_launch(void* const* d_in, const int* in_sizes, int n_in,
                              void* d_out, int out_size, void* d_ws, size_t ws_size,
                              hipStream_t stream) {
  (void)in_sizes; (void)n_in; (void)out_size; (void)d_ws; (void)ws_size;
  const float* x  = (const float*)d_in[0];
  const float* W1 = (const float*)d_in[1];
  const float* U1 = (const float*)d_in[2];
  const float* b1 = (const float*)d_in[3];
  const float* W2 = (const float*)d_in[4];
  const float* U2 = (const float*)d_in[5];
  const float* b2 = (const float*)d_in[6];
  const float* W3 = (const float*)d_in[7];
  const float* U3 = (const float*)d_in[8];
  const float* b3 = (const float*)d_in[9];
  const float* W4 = (const float*)d_in[10];
  const float* U4 = (const float*)d_in[11];
  const float* b4 = (const float*)d_in[12];
  const float* Wd = (const float*)d_in[13];
  const float* bd = (const float*)d_in[14];
  float* out = (float*)d_out;

  dim3 grid(BATCH / 16), block(256);
  hipLaunchKernelGGL(lstm_ae_kernel, grid, block, 0, stream,
                     x, W1, U1, b1, W2, U2, b2, W3, U3, b3, W4, U4, b4, Wd, bd, out);
}